// CompositeHeadB_47485158425160
// MI455X (gfx1250) — compile-verified
//
#include <hip/hip_runtime.h>

typedef float v2f __attribute__((ext_vector_type(2)));
typedef float v8f __attribute__((ext_vector_type(8)));

#define E        256
#define V1       17
#define T_LAT    12872
#define T_TOK    74312
#define NBATCH   4

// LAT offsets (rows of x) and token offsets (rows of out)
#define XOFF3    584      // seg index 3 (K4 segment) x-row base
#define XOFF4    4680     // seg index 4 (K5 segment) x-row base
#define TOK3     584      // token base of segment 3
#define TOK4     8776     // token base of segment 4
#define LAT3     4096
#define LAT4     8192

// workspace layout (floats): 13 matrices of 256x32 (pair-interleaved) then 13 biases of 32
#define MAT_STRIDE   8192         // 256*32
#define BIAS_OFF     (13*8192)
#define LDS_STRIDE   260          // dwords per staged row (even, 4*65 -> conflict-free b64)

// M layout: element (i, o) stored at M[(i>>1)*64 + o*2 + (i&1)]
__device__ __forceinline__ int midx(int i, int o) { return ((i >> 1) << 6) + (o << 1) + (i & 1); }

// ---------------------------------------------------------------------------
// Kernel 1: build fused matrices M_k = K[:,:,k] @ W (or plain W for segs 0-2)
// and fused biases c = cb@W + b, in WMMA-B-friendly interleaved layout.
// ---------------------------------------------------------------------------
__global__ __launch_bounds__(256) void precompute(
    const float* __restrict__ W1, const float* __restrict__ b1,
    const float* __restrict__ W2, const float* __restrict__ b2,
    const float* __restrict__ W3, const float* __restrict__ b3,
    const float* __restrict__ K4, const float* __restrict__ cb4,
    const float* __restrict__ W4, const float* __restrict__ b4,
    const float* __restrict__ K5, const float* __restrict__ cb5,
    const float* __restrict__ W5, const float* __restrict__ b5,
    float* __restrict__ ws)
{
    const int m   = blockIdx.x;          // 0..12
    const int tid = threadIdx.x;         // 0..255
    float* M = ws + (size_t)m * MAT_STRIDE;
    float* c = ws + BIAS_OFF + m * 32;

    if (m < 3) {
        const float* W = (m == 0) ? W1 : (m == 1) ? W2 : W3;
        const float* b = (m == 0) ? b1 : (m == 1) ? b2 : b3;
        const int i = tid;
        for (int o = 0; o < 32; ++o)
            M[midx(i, o)] = (o < V1) ? W[i * V1 + o] : 0.0f;
        if (tid < 32) c[tid] = (tid < V1) ? b[tid] : 0.0f;
        return;
    }

    const bool is4 = (m < 5);
    const float* K  = is4 ? K4  : K5;
    const float* W  = is4 ? W4  : W5;
    const float* b  = is4 ? b4  : b5;
    const float* cb = is4 ? cb4 : cb5;
    const int nk = is4 ? 2 : 8;
    const int kk = is4 ? (m - 3) : (m - 5);

    const int i = tid;                   // input-dim row of M
    float acc[V1];
    #pragma unroll
    for (int v = 0; v < V1; ++v) acc[v] = 0.0f;
    for (int e = 0; e < E; ++e) {        // contraction over K's middle dim
        const float kv = K[((size_t)i * E + e) * nk + kk];
        #pragma unroll
        for (int v = 0; v < V1; ++v) acc[v] = fmaf(kv, W[e * V1 + v], acc[v]);
    }
    for (int o = 0; o < 32; ++o)
        M[midx(i, o)] = (o < V1) ? acc[o] : 0.0f;

    if (tid < 32) {
        if (tid < V1) {
            float bacc = b[tid];
            for (int e = 0; e < E; ++e) bacc = fmaf(cb[e], W[e * V1 + tid], bacc);
            c[tid] = bacc;
        } else c[tid] = 0.0f;
    }
}

// ---------------------------------------------------------------------------
// Kernel 2: tiny segments 0-2 (584 tokens/batch) — plain VALU dot products.
// ---------------------------------------------------------------------------
__global__ __launch_bounds__(64) void head_small(
    const float* __restrict__ x, const float* __restrict__ ws, float* __restrict__ out)
{
    const int gid = blockIdx.x * blockDim.x + threadIdx.x;
    const int total = NBATCH * 584 * V1;
    if (gid >= total) return;
    const int o = gid % V1;
    const int j = (gid / V1) % 584;
    const int n = gid / (V1 * 584);
    const int m = (j < 8) ? 0 : (j < 72) ? 1 : 2;

    const float* M = ws + (size_t)m * MAT_STRIDE;
    const float* c = ws + BIAS_OFF + m * 32;
    const float* xr = x + ((size_t)n * T_LAT + j) * E;

    float acc = c[o];
    #pragma unroll 4
    for (int i = 0; i < E; ++i) acc = fmaf(xr[i], M[midx(i, o)], acc);
    out[((size_t)n * T_TOK + j) * V1 + o] = acc;
}

// ---------------------------------------------------------------------------
// Kernel 3: segments 3-4 via V_WMMA_F32_16X16X4_F32.
// One wave handles one 16-row x tile for all its k-slices (A staged in LDS).
// Output cols 0-15 via WMMA; col 16 via VALU + shfl_xor(16) reduction.
// ---------------------------------------------------------------------------
#define WAVES 2
__global__ __launch_bounds__(64) void head_wmma(
    const float* __restrict__ x, const float* __restrict__ ws, float* __restrict__ out)
{
    __shared__ float lds[WAVES * 16 * LDS_STRIDE];
    const int tid  = threadIdx.x;
    const int wave = tid >> 5;
    const int lane = tid & 31;

    const int w   = blockIdx.x * WAVES + wave;   // 0..3071, exact fit
    const int n   = w / 768;
    const int rem = w % 768;

    int t0, nk, matbase, xrow, tokbase;
    if (rem < 256) { t0 = rem * 16;          nk = 2; matbase = 3; xrow = XOFF3 + t0; tokbase = TOK3; }
    else           { t0 = (rem - 256) * 16;  nk = 8; matbase = 5; xrow = XOFF4 + t0; tokbase = TOK4; }

    float* myl = lds + wave * 16 * LDS_STRIDE;
    const float* xb = x + ((size_t)n * T_LAT + xrow) * E;

    // stage 16x256 fp32 A tile into LDS (float4 coalesced)
    for (int idx = lane; idx < 16 * 64; idx += 32) {
        const int r  = idx >> 6;
        const int c4 = idx & 63;
        const float4 v = *(const float4*)(xb + (size_t)r * E + c4 * 4);
        *(float4*)(myl + r * LDS_STRIDE + c4 * 4) = v;
    }
    __syncthreads();

    const int hi  = lane >> 4;      // lane group (selects K sub-pair)
    const int col = lane & 15;      // A row (M) for loads; B/D column (N)
    const float* arow = myl + col * LDS_STRIDE;
    const float* biasbase = ws + BIAS_OFF;

    for (int k = 0; k < nk; ++k) {
        const float* Mk = ws + (size_t)(matbase + k) * MAT_STRIDE;
        const float* ck = biasbase + (matbase + k) * 32;

        const float bc = ck[col];
        v8f c0;
        #pragma unroll
        for (int r = 0; r < 8; ++r) c0[r] = bc;   // D = A@M + bias
        float acc16 = 0.0f;                        // column 16 partial

        #pragma unroll 4
        for (int i = 0; i < E; i += 4) {
            // A frag: lanes 0-15 / 16-31 hold M=0-15; VGPR0/1 = K {i+2hi, i+2hi+1}
            const v2f a = *(const v2f*)(arow + i + 2 * hi);
            // B frag: rows {i+2hi, i+2hi+1} at column `col` (pair-interleaved layout)
            const float* mb = Mk + (size_t)((i >> 1) + hi) * 64;
            const v2f b0 = *(const v2f*)(mb + col * 2);
            c0 = __builtin_amdgcn_wmma_f32_16x16x4_f32(
                     false, a, false, b0, (short)0, c0, false, false);
            // column 16 via VALU on the same A frag
            const v2f b16 = *(const v2f*)(mb + 32);   // o = 16
            acc16 = fmaf(a.x, b16.x, fmaf(a.y, b16.y, acc16));
        }

        // reduce col-16 partials across the two K lane-groups (row = col in both)
        acc16 += __shfl_xor(acc16, 16, 32);

        // store WMMA tile: lane holds (M = r + 8*hi, N = col)
        #pragma unroll
        for (int r = 0; r < 8; ++r) {
            const int mrow = r + 8 * hi;
            const size_t jrow = (size_t)tokbase + (size_t)(t0 + mrow) * nk + k;
            float* op = out + ((size_t)n * T_TOK + jrow) * V1;
            op[col] = c0[r];
        }
        if (hi == 0) {  // lanes 0-15 hold full col-16 dot for rows 0-15
            const size_t jrow = (size_t)tokbase + (size_t)(t0 + col) * nk + k;
            out[((size_t)n * T_TOK + jrow) * V1 + 16] = acc16 + ck[16];
        }
    }
}

// ---------------------------------------------------------------------------
extern "C" void kernel_launch(void* const* d_in, const int* in_sizes, int n_in,
                              void* d_out, int out_size, void* d_ws, size_t ws_size,
                              hipStream_t stream)
{
    const float* x   = (const float*)d_in[0];
    // d_in[1..3] = value, depth, position (unused by the forward pass)
    const float* W1  = (const float*)d_in[4];
    const float* b1  = (const float*)d_in[5];
    const float* W2  = (const float*)d_in[6];
    const float* b2  = (const float*)d_in[7];
    const float* W3  = (const float*)d_in[8];
    const float* b3  = (const float*)d_in[9];
    const float* K4  = (const float*)d_in[10];
    const float* cb4 = (const float*)d_in[11];
    const float* W4  = (const float*)d_in[12];
    const float* b4  = (const float*)d_in[13];
    const float* K5  = (const float*)d_in[14];
    const float* cb5 = (const float*)d_in[15];
    const float* W5  = (const float*)d_in[16];
    const float* b5  = (const float*)d_in[17];
    float* out = (float*)d_out;
    float* ws  = (float*)d_ws;   // needs 13*8192 + 13*32 floats (~418 KB)

    precompute<<<13, 256, 0, stream>>>(W1, b1, W2, b2, W3, b3,
                                       K4, cb4, W4, b4, K5, cb5, W5, b5, ws);

    const int small_total = NBATCH * 584 * V1;
    head_small<<<(small_total + 63) / 64, 64, 0, stream>>>(x, ws, out);

    // 3072 waves total: 4 batches * (256 seg3-tiles + 512 seg4-tiles), 2 waves/block
    head_wmma<<<3072 / WAVES, 32 * WAVES, 0, stream>>>(x, ws, out);
}